// SpikingWorkingMemory_67018669686945
// MI455X (gfx1250) — compile-verified
//
#include <hip/hip_runtime.h>

// ----------------------------------------------------------------------------
// SpikingWorkingMemory on MI455X (gfx1250), persistent single-kernel design.
//
//  * W_rec is block-diagonal (32 x [256x256]); the actual block data is kept
//    LDS-resident for all 64 substeps (256KB block <= 320KB WGP LDS), loaded
//    via GLOBAL_LOAD_ASYNC_TO_LDS_B128 when available.
//  * Block matvec via V_WMMA_F32_16X16X4_F32 (full fp32 -> safe at the v>=30
//    spike threshold). B operand = re broadcast across N; D columns identical.
//  * W_inh / W_inh_back rows are constant -> (W@x)[i] = W[i,0] * sum(x).
//  * Inhibitory population (2048 neurons) is simulated REDUNDANTLY in every
//    workgroup (8 neurons/thread): its only coupling to the rest of the system
//    is the scalar sum(re), so each WG derives sum(ri) locally and bit-
//    identically. This cuts global barriers to 1/substep (63 total; the last
//    substep's inhibitory update is dead code).
//  * sum(re) exchanged via device-scope atomics + one-shot counters zeroed by
//    hipMemsetAsync (graph-capture safe, deterministic across replays).
// ----------------------------------------------------------------------------

typedef float v2f __attribute__((ext_vector_type(2)));
typedef float v8f __attribute__((ext_vector_type(8)));

#define N_EXC    8192
#define N_INH    2048
#define N_SLOTS  32
#define NPS      256   // exc neurons per slot
#define INH_PT   8     // inh neurons per thread (256 threads -> all 2048)
#define SUBSTEPS 64

// LDS layout (float offsets)
#define W_OFF      0
#define RE_OFF     (256 * 256)
#define Y_OFF      (RE_OFF + 256)
#define RED_OFF    (Y_OFF + 256)
#define RSUM_OFF   (RED_OFF + 256)
#define LDS_FLOATS (RSUM_OFF + 32)

// workspace layout (bytes)
#define WS_COUNTERS_BYTES 1024          // 63 counters used, zeroed per launch
#define WS_REPART_OFF     1024          // float[SUBSTEPS][32]

#ifndef __has_builtin
#define __has_builtin(x) 0
#endif
#if defined(__HIP_DEVICE_COMPILE__) && \
    __has_builtin(__builtin_amdgcn_global_load_async_to_lds_b128) && \
    __has_builtin(__builtin_amdgcn_s_wait_asynccnt)
#define USE_ASYNC_PRELOAD 1
#else
#define USE_ASYNC_PRELOAD 0
#endif

#if USE_ASYNC_PRELOAD
// Parameter types per clang's diagnostic: pointer to 4-int vector, AS1 source
// (global) and AS3 destination (LDS).
typedef int v4i __attribute__((vector_size(16)));
typedef __attribute__((address_space(1))) v4i* as1_v4i_p;
typedef __attribute__((address_space(3))) v4i* as3_v4i_p;
#endif

__device__ __forceinline__ void gbar(unsigned int* counters, int b) {
    __syncthreads();
    if (threadIdx.x == 0) {
        __hip_atomic_fetch_add(&counters[b], 1u, __ATOMIC_RELEASE,
                               __HIP_MEMORY_SCOPE_AGENT);
        while (__hip_atomic_load(&counters[b], __ATOMIC_ACQUIRE,
                                 __HIP_MEMORY_SCOPE_AGENT) < (unsigned)gridDim.x) {
            __builtin_amdgcn_s_sleep(1);
        }
    }
    __syncthreads();
}

__global__ void __launch_bounds__(256)
snn_persistent_kernel(const float* __restrict__ input_drive,
                      const float* __restrict__ W_rec,
                      const float* __restrict__ W_inh,
                      const float* __restrict__ W_inh_back,
                      const float* __restrict__ v_exc0,
                      const float* __restrict__ u_exc0,
                      const float* __restrict__ r_exc0,
                      const float* __restrict__ v_inh0,
                      const float* __restrict__ u_inh0,
                      const float* __restrict__ r_inh0,
                      const float* __restrict__ noise_exc,
                      const float* __restrict__ noise_inh,
                      float* __restrict__ out,
                      unsigned int* __restrict__ counters,
                      float* __restrict__ re_part)   // [SUBSTEPS][32]
{
    extern __shared__ float smem[];
    float* Wl   = smem + W_OFF;     // 256x256 fp32 block, row major
    float* re_l = smem + RE_OFF;    // current exc rates of this slot
    float* y_l  = smem + Y_OFF;     // matvec result
    float* red  = smem + RED_OFF;   // reduction scratch (256)
    float* rsum = smem + RSUM_OFF;  // 32 partials staging

    const int tid  = threadIdx.x;
    const int slot = blockIdx.x;
    const int g    = slot * NPS + tid;   // global exc index

    // ---- load this slot's W block into LDS (one-time) ----
    {
        const float* wbase = W_rec + (size_t)(slot * NPS) * N_EXC + (size_t)slot * NPS;
#if USE_ASYNC_PRELOAD
        for (int i = tid; i < 256 * 64; i += 256) {          // 16384 x b128
            int r = i >> 6, c4 = (i & 63) << 2;
            __builtin_amdgcn_global_load_async_to_lds_b128(
                (as1_v4i_p)(wbase + (size_t)r * N_EXC + c4),
                (as3_v4i_p)(&Wl[r * 256 + c4]), 0, 0);
        }
        __builtin_amdgcn_s_wait_asynccnt(0);
#else
        for (int i = tid; i < 256 * 64; i += 256) {          // 16384 float4
            int r = i >> 6, c4 = (i & 63) << 2;
            float4 val = *reinterpret_cast<const float4*>(wbase + (size_t)r * N_EXC + c4);
            *reinterpret_cast<float4*>(&Wl[r * 256 + c4]) = val;
        }
#endif
    }

    // ---- init exc state (1 neuron / thread) ----
    const float I_ext    = input_drive[g] * 8.0f;
    const float coef_i2e = W_inh_back[(size_t)g * N_INH] * 8.0f;   // row-constant
    float v = v_exc0[g], u = u_exc0[g], re = r_exc0[g];
    re_l[tid] = re;

    // ---- init inh state: this WG simulates ALL 2048 inh neurons (8/thread) ----
    float viA[INH_PT], uiA[INH_PT], riA[INH_PT], ce2i[INH_PT];
    {
        float lsum = 0.0f;
        #pragma unroll
        for (int k = 0; k < INH_PT; ++k) {
            int j = tid + k * 256;                       // global inh index
            viA[k]  = v_inh0[j];
            uiA[k]  = u_inh0[j];
            riA[k]  = r_inh0[j];
            ce2i[k] = W_inh[(size_t)j * N_EXC] * 5.0f;   // row-constant
            lsum += riA[k];
        }
        red[tid] = lsum;
    }
    __syncthreads();   // also fences W preload + re_l init for all waves
    for (int s = 128; s > 0; s >>= 1) { if (tid < s) red[tid] += red[tid + s]; __syncthreads(); }
    float sri = red[0];            // global sum(ri), identical in every WG

    // ---- WMMA lane geometry (fp32 16x16x4) ----
    const int lane = tid & 31;
    const int wave = tid >> 5;
    const int kl   = (lane >> 4) << 1;            // k base within chunk: 0 or 2
    const int r0   = wave * 32 + (lane & 15);     // rowgroup 2*wave
    const int r1   = r0 + 16;                     // rowgroup 2*wave + 1

    float wg_re_sum = 0.0f;

    for (int t = 0; t < SUBSTEPS; ++t) {
        // --- y = W_block @ re via V_WMMA_F32_16X16X4_F32 (64 K-chunks) ---
        v8f c0 = {0, 0, 0, 0, 0, 0, 0, 0};
        v8f c1 = {0, 0, 0, 0, 0, 0, 0, 0};
        const float* arow0 = &Wl[r0 * 256 + kl];
        const float* arow1 = &Wl[r1 * 256 + kl];
        const float* bp    = &re_l[kl];
        #pragma unroll 4
        for (int kc = 0; kc < 64; ++kc) {
            v2f a0 = *reinterpret_cast<const v2f*>(arow0 + kc * 4);
            v2f a1 = *reinterpret_cast<const v2f*>(arow1 + kc * 4);
            v2f bb = *reinterpret_cast<const v2f*>(bp + kc * 4);   // broadcast across N
            c0 = __builtin_amdgcn_wmma_f32_16x16x4_f32(false, a0, false, bb, (short)0, c0, false, false);
            c1 = __builtin_amdgcn_wmma_f32_16x16x4_f32(false, a1, false, bb, (short)0, c1, false, false);
        }
        // D columns identical (B broadcast): lanes with N==0 (0 and 16) extract.
        if ((lane & 15) == 0) {
            int half8 = (lane >> 4) * 8;   // vgpr p: lanes 0-15 -> M=p, lanes 16-31 -> M=p+8
            #pragma unroll
            for (int p = 0; p < 8; ++p) {
                y_l[wave * 32 + half8 + p]      = c0[p];
                y_l[wave * 32 + 16 + half8 + p] = c1[p];
            }
        }
        __syncthreads();

        // --- excitatory Izhikevich step (RS: a=.02 b=.2 c=-65 d=8) ---
        {
            float Irec  = y_l[tid] * 10.0f;
            float ne    = noise_exc[t * N_EXC + g];
            float Itot  = I_ext + Irec + coef_i2e * sri + ne * 0.3f + 1.0f;  // + I_TONIC
            float v_new = v + (0.04f * v * v + 5.0f * v + 140.0f - u + Itot);
            float u_new = u + 0.02f * (0.2f * v - u);
            bool  sp    = (v_new >= 30.0f);
            v  = sp ? -65.0f : v_new;
            u  = u_new + (sp ? 8.0f : 0.0f);
            re = re * 0.9f + (sp ? 0.1f : 0.0f);
        }
        re_l[tid] = re;          // safe: matvec reads completed before y sync
        red[tid]  = re;
        __syncthreads();
        for (int s = 128; s > 0; s >>= 1) { if (tid < s) red[tid] += red[tid + s]; __syncthreads(); }
        if (tid == 0) wg_re_sum = red[0];

        if (t == SUBSTEPS - 1) break;   // final inhibitory update is dead code

        if (tid == 0)
            __hip_atomic_store(&re_part[t * 32 + slot], red[0], __ATOMIC_RELEASE,
                               __HIP_MEMORY_SCOPE_AGENT);
        gbar(counters, t);              // one global barrier per substep

        // --- global sum(re) from the 32 slot partials ---
        if (tid < 32)
            rsum[tid] = __hip_atomic_load(&re_part[t * 32 + tid], __ATOMIC_ACQUIRE,
                                          __HIP_MEMORY_SCOPE_AGENT);
        __syncthreads();
        float sre = 0.0f;
        #pragma unroll
        for (int i = 0; i < 32; ++i) sre += rsum[i];

        // --- inhibitory Izhikevich step (FS: a=.1 b=.2 c=-65 d=2), replicated ---
        {
            float lsum = 0.0f;
            #pragma unroll
            for (int k = 0; k < INH_PT; ++k) {
                float ni    = noise_inh[t * N_INH + tid + k * 256];
                float Itot  = ce2i[k] * sre + ni * 0.3f + 1.0f;
                float vk    = viA[k], uk = uiA[k];
                float v_new = vk + (0.04f * vk * vk + 5.0f * vk + 140.0f - uk + Itot);
                float u_new = uk + 0.10f * (0.2f * vk - uk);
                bool  sp    = (v_new >= 30.0f);
                viA[k] = sp ? -65.0f : v_new;
                uiA[k] = u_new + (sp ? 2.0f : 0.0f);
                riA[k] = riA[k] * 0.9f + (sp ? 0.1f : 0.0f);
                lsum  += riA[k];
            }
            red[tid] = lsum;
        }
        __syncthreads();
        for (int s = 128; s > 0; s >>= 1) { if (tid < s) red[tid] += red[tid + s]; __syncthreads(); }
        sri = red[0];                   // next substep's global sum(ri)
    }

    // ---- outputs: exc_rate (8192) then slot_activity (32) ----
    out[g] = re;
    if (tid == 0) out[N_EXC + slot] = wg_re_sum * (1.0f / 256.0f);
}

extern "C" void kernel_launch(void* const* d_in, const int* in_sizes, int n_in,
                              void* d_out, int out_size, void* d_ws, size_t ws_size,
                              hipStream_t stream) {
    (void)in_sizes; (void)n_in; (void)out_size; (void)ws_size;

    const float* input_drive = (const float*)d_in[0];
    const float* W_rec       = (const float*)d_in[1];
    const float* W_inh       = (const float*)d_in[2];
    const float* W_inh_back  = (const float*)d_in[3];
    const float* v_exc       = (const float*)d_in[4];
    const float* u_exc       = (const float*)d_in[5];
    const float* r_exc       = (const float*)d_in[6];
    const float* v_inh       = (const float*)d_in[7];
    const float* u_inh       = (const float*)d_in[8];
    const float* r_inh       = (const float*)d_in[9];
    const float* noise_exc   = (const float*)d_in[10];
    const float* noise_inh   = (const float*)d_in[11];
    float* out = (float*)d_out;

    char* ws = (char*)d_ws;
    unsigned int* counters = (unsigned int*)ws;
    float* re_part = (float*)(ws + WS_REPART_OFF);

    // Zero the one-shot barrier counters every call (graph-capturable).
    (void)hipMemsetAsync(d_ws, 0, WS_COUNTERS_BYTES, stream);

    const size_t lds_bytes = (size_t)LDS_FLOATS * sizeof(float);  // ~259 KB < 320 KB WGP LDS
    (void)hipFuncSetAttribute((const void*)snn_persistent_kernel,
                              hipFuncAttributeMaxDynamicSharedMemorySize, (int)lds_bytes);

    snn_persistent_kernel<<<dim3(N_SLOTS), dim3(256), lds_bytes, stream>>>(
        input_drive, W_rec, W_inh, W_inh_back,
        v_exc, u_exc, r_exc, v_inh, u_inh, r_inh,
        noise_exc, noise_inh, out, counters, re_part);
}